// ConstrainedSparseClusterDecomposition_120259084485
// MI455X (gfx1250) — compile-verified
//
#include <hip/hip_runtime.h>
#include <hip/hip_bf16.h>

// ---------------- problem constants (from reference) ----------------
#define BB   16
#define NN   8192
#define DD   128
#define CC   64
#define KK   4
#define ROWS (BB * NN)            // 131072 total rows (B*N)
#define TEMP 6.0f                  // 2.0 * (1 + 192/96)
#define INV_TEMP (1.0f / 6.0f)

// kernel 1 geometry
#define WAVES1   4
#define BLOCK1   (WAVES1 * 32)     // 128 threads
#define ROWS_PB  (WAVES1 * 16)     // 64 rows per block
#define GRID1    (ROWS / ROWS_PB)  // 2048 blocks

// KL kernel geometry
#define BLOCKC 256
#define GRIDC  (ROWS / BLOCKC)     // 512 blocks

// workspace layout (float offsets)
#define WS_SCORES   ((size_t)0)                          // ROWS*CC = 8388608
#define WS_COLPART  ((size_t)ROWS * CC)                  // GRID1*CC = 131072
#define WS_F        (WS_COLPART + (size_t)GRID1 * CC)    // 64
#define WS_ORTHO    (WS_F + CC)                          // 1
#define WS_KLPART   (WS_ORTHO + 1)                       // GRIDC = 512

typedef float v2f __attribute__((ext_vector_type(2)));
typedef float v8f __attribute__((ext_vector_type(8)));

// --------------------------------------------------------------------
// Kernel 1: fused scores GEMM (WMMA f32 16x16x4) + softmax + top-4
//           routing + x_common / x_residual, stage scores + colsum
// --------------------------------------------------------------------
__global__ __launch_bounds__(BLOCK1)
void csc_main_kernel(const float* __restrict__ x,
                     const float* __restrict__ dict,
                     float* __restrict__ out_common,
                     float* __restrict__ out_resid,
                     float* __restrict__ scores_ws,
                     float* __restrict__ colsum_part)
{
    __shared__ float sDict[CC * DD];              // 32 KB, row-major [c][d]
    __shared__ float sScore[WAVES1][16][CC];      // 16 KB, later holds q
    __shared__ int   sTopIdx[WAVES1][16][KK];
    __shared__ float sTopW[WAVES1][16][KK];

    const int tid  = threadIdx.x;
    const int wave = tid >> 5;
    const int lane = tid & 31;

    // ---- load dictionary into LDS (coalesced float4) ----
    {
        const float4* src = reinterpret_cast<const float4*>(dict);
        float4*       dst = reinterpret_cast<float4*>(sDict);
        for (int i = tid; i < (CC * DD) / 4; i += BLOCK1) dst[i] = src[i];
    }
    __syncthreads();

    const int rowBase = blockIdx.x * ROWS_PB + wave * 16;

    // ---- WMMA: this wave computes a 16(M) x 64(C) score tile ----
    // A (16x4 f32): lanes 0-15 -> M=lane, K0/K1; lanes 16-31 -> M=lane-16, K2/K3
    // B (4x16 f32): lanes 0-15 -> N=lane, K0/K1; lanes 16-31 -> N=lane-16, K2/K3
    v8f acc0 = {}, acc1 = {}, acc2 = {}, acc3 = {};
    const int mn   = lane & 15;          // M for A, N (local c) for B
    const int kOff = (lane >> 4) * 2;    // 0 or 2
    const float* xrow = x + (size_t)(rowBase + mn) * DD;

    for (int kb = 0; kb < DD; kb += 4) {
        v2f a;
        a.x = xrow[kb + kOff];
        a.y = xrow[kb + kOff + 1];

        v2f b0, b1, b2, b3;
        b0.x = sDict[(mn      ) * DD + kb + kOff];
        b0.y = sDict[(mn      ) * DD + kb + kOff + 1];
        b1.x = sDict[(mn + 16 ) * DD + kb + kOff];
        b1.y = sDict[(mn + 16 ) * DD + kb + kOff + 1];
        b2.x = sDict[(mn + 32 ) * DD + kb + kOff];
        b2.y = sDict[(mn + 32 ) * DD + kb + kOff + 1];
        b3.x = sDict[(mn + 48 ) * DD + kb + kOff];
        b3.y = sDict[(mn + 48 ) * DD + kb + kOff + 1];

        acc0 = __builtin_amdgcn_wmma_f32_16x16x4_f32(false, a, false, b0, (short)0, acc0, false, false);
        acc1 = __builtin_amdgcn_wmma_f32_16x16x4_f32(false, a, false, b1, (short)0, acc1, false, false);
        acc2 = __builtin_amdgcn_wmma_f32_16x16x4_f32(false, a, false, b2, (short)0, acc2, false, false);
        acc3 = __builtin_amdgcn_wmma_f32_16x16x4_f32(false, a, false, b3, (short)0, acc3, false, false);
    }

    // ---- spill accumulators to LDS (C/D layout: M = vgpr + 8*(lane>>4), N = lane&15) ----
    {
        const int rhalf = (lane >> 4) * 8;
#pragma unroll
        for (int j = 0; j < 8; ++j) {
            sScore[wave][rhalf + j][ 0 + mn] = acc0[j];
            sScore[wave][rhalf + j][16 + mn] = acc1[j];
            sScore[wave][rhalf + j][32 + mn] = acc2[j];
            sScore[wave][rhalf + j][48 + mn] = acc3[j];
        }
    }
    __syncthreads();

    // ---- stage raw scores to workspace (coalesced; LDS is contiguous [64][64]) ----
    {
        const float* sFlat = &sScore[0][0][0];
        const size_t g0 = (size_t)blockIdx.x * ROWS_PB * CC;
        for (int i = tid; i < ROWS_PB * CC; i += BLOCK1)
            scores_ws[g0 + i] = sFlat[i];
    }
    __syncthreads();

    // ---- per-row: softmax stats, top-4 routing, overwrite LDS with q ----
    if (lane < 16) {
        const int r = lane;
        float* srow = sScore[wave][r];

        float mmax = srow[0];
#pragma unroll
        for (int c = 1; c < CC; ++c) mmax = fmaxf(mmax, srow[c]);

        float sumexp = 0.0f;
#pragma unroll
        for (int c = 0; c < CC; ++c) sumexp += __expf((srow[c] - mmax) * INV_TEMP);

        // top-4 (strict >, first index wins -> matches lax.top_k tie order)
        int   idx[KK];
        float val[KK];
#pragma unroll
        for (int k = 0; k < KK; ++k) {
            float best = -3.4e38f;
            int   bi   = 0;
            for (int c = 0; c < CC; ++c) {
                bool used = false;
                for (int t = 0; t < k; ++t) used = used || (idx[t] == c);
                const float v = srow[c];
                if (!used && v > best) { best = v; bi = c; }
            }
            idx[k] = bi;
            val[k] = best;
        }
        // softmax over top-4 / temp (val[0] is the max)
        float wsum = 0.0f, wv[KK];
#pragma unroll
        for (int k = 0; k < KK; ++k) { wv[k] = __expf((val[k] - val[0]) * INV_TEMP); wsum += wv[k]; }
#pragma unroll
        for (int k = 0; k < KK; ++k) {
            sTopW[wave][r][k]   = wv[k] / wsum;
            sTopIdx[wave][r][k] = idx[k];
        }

        // overwrite scores with q (raw scores already staged to ws)
        const float inv = 1.0f / sumexp;
#pragma unroll
        for (int c = 0; c < CC; ++c) srow[c] = __expf((srow[c] - mmax) * INV_TEMP) * inv;
    }
    __syncthreads();

    // ---- deterministic per-block column sums of q ----
    if (tid < CC) {
        const float* sFlat = &sScore[0][0][0];
        float s = 0.0f;
        for (int r = 0; r < ROWS_PB; ++r) s += sFlat[r * CC + tid];
        colsum_part[(size_t)blockIdx.x * CC + tid] = s;
    }

    // ---- x_common = sum_k w_k * dict[idx_k]; residual = x - x_common ----
    {
        const int d0 = lane * 4;   // 32 lanes * 4 = 128 = D
        for (int r = 0; r < 16; ++r) {
            const size_t row = (size_t)(rowBase + r);
            const float w0 = sTopW[wave][r][0], w1 = sTopW[wave][r][1];
            const float w2 = sTopW[wave][r][2], w3 = sTopW[wave][r][3];
            const int   i0 = sTopIdx[wave][r][0], i1 = sTopIdx[wave][r][1];
            const int   i2 = sTopIdx[wave][r][2], i3 = sTopIdx[wave][r][3];

            float xc[4];
#pragma unroll
            for (int j = 0; j < 4; ++j) {
                const int d = d0 + j;
                xc[j] = w0 * sDict[i0 * DD + d] + w1 * sDict[i1 * DD + d]
                      + w2 * sDict[i2 * DD + d] + w3 * sDict[i3 * DD + d];
            }
            const float4 xv = *reinterpret_cast<const float4*>(x + row * DD + d0);
            float4 oc, orr;
            oc.x = xc[0]; oc.y = xc[1]; oc.z = xc[2]; oc.w = xc[3];
            orr.x = xv.x - xc[0]; orr.y = xv.y - xc[1];
            orr.z = xv.z - xc[2]; orr.w = xv.w - xc[3];
            *reinterpret_cast<float4*>(out_common + row * DD + d0) = oc;
            *reinterpret_cast<float4*>(out_resid  + row * DD + d0) = orr;
        }
    }
}

// --------------------------------------------------------------------
// Kernel 2: reduce column-sum partials -> f[64]; ortho loss sum
// --------------------------------------------------------------------
__global__ __launch_bounds__(256)
void csc_reduce_kernel(const float* __restrict__ colsum_part,
                       const float* __restrict__ dict,
                       float* __restrict__ f,
                       float* __restrict__ ortho)
{
    __shared__ float red[256];
    const int tid = threadIdx.x;

    if (tid < CC) {
        float s = 0.0f;
        for (int b = 0; b < GRID1; ++b) s += colsum_part[(size_t)b * CC + tid];
        f[tid] = s;
    }

    // ortho: sum((dict*dict^T - I)^2) over 64x64 entries
    float o = 0.0f;
    for (int e = tid; e < CC * CC; e += 256) {
        const int i = e >> 6, j = e & 63;
        float dot = 0.0f;
        for (int d = 0; d < DD; ++d) dot += dict[i * DD + d] * dict[j * DD + d];
        const float diff = dot - ((i == j) ? 1.0f : 0.0f);
        o += diff * diff;
    }
    red[tid] = o;
    __syncthreads();
    for (int s = 128; s > 0; s >>= 1) {
        if (tid < s) red[tid] += red[tid + s];
        __syncthreads();
    }
    if (tid == 0) *ortho = red[0];
}

// --------------------------------------------------------------------
// Kernel 3: KL(p || q) partial sums; one row per thread
// --------------------------------------------------------------------
__global__ __launch_bounds__(BLOCKC)
void csc_kl_kernel(const float* __restrict__ scores_ws,
                   const float* __restrict__ f,
                   float* __restrict__ kl_part)
{
    __shared__ float sF[CC];
    __shared__ float red[BLOCKC];
    const int tid = threadIdx.x;
    if (tid < CC) sF[tid] = f[tid];
    __syncthreads();

    const size_t row = (size_t)blockIdx.x * BLOCKC + tid;
    const float* srow = scores_ws + row * CC;

    float sv[CC];
#pragma unroll
    for (int c = 0; c < CC; ++c) sv[c] = srow[c];

    float mmax = sv[0];
#pragma unroll
    for (int c = 1; c < CC; ++c) mmax = fmaxf(mmax, sv[c]);

    float sumexp = 0.0f;
#pragma unroll
    for (int c = 0; c < CC; ++c) sumexp += __expf((sv[c] - mmax) * INV_TEMP);
    const float lse = __logf(sumexp);

    float W = 0.0f;
#pragma unroll
    for (int c = 0; c < CC; ++c) {
        const float lq = (sv[c] - mmax) * INV_TEMP - lse;
        const float q  = __expf(lq);
        W += q * q / sF[c];
    }

    float kl = 0.0f;
#pragma unroll
    for (int c = 0; c < CC; ++c) {
        const float lq = (sv[c] - mmax) * INV_TEMP - lse;
        const float q  = __expf(lq);
        const float p  = (q * q / sF[c]) / W;
        kl += p * (__logf(p) - lq);
    }

    red[tid] = kl;
    __syncthreads();
    for (int s = BLOCKC / 2; s > 0; s >>= 1) {
        if (tid < s) red[tid] += red[tid + s];
        __syncthreads();
    }
    if (tid == 0) kl_part[blockIdx.x] = red[0];
}

// --------------------------------------------------------------------
// Kernel 4: finalize aux_loss
// --------------------------------------------------------------------
__global__ __launch_bounds__(256)
void csc_final_kernel(const float* __restrict__ kl_part,
                      const float* __restrict__ ortho,
                      float* __restrict__ aux_out)
{
    __shared__ float red[256];
    const int tid = threadIdx.x;
    red[tid] = kl_part[tid] + kl_part[tid + 256];   // GRIDC == 512
    __syncthreads();
    for (int s = 128; s > 0; s >>= 1) {
        if (tid < s) red[tid] += red[tid + s];
        __syncthreads();
    }
    if (tid == 0) {
        const float kl = red[0] / (float)ROWS;      // batchmean
        aux_out[0] = kl * 0.5f + 0.1f * ((*ortho) / (float)(CC * CC));
    }
}

// --------------------------------------------------------------------
extern "C" void kernel_launch(void* const* d_in, const int* in_sizes, int n_in,
                              void* d_out, int out_size, void* d_ws, size_t ws_size,
                              hipStream_t stream) {
    (void)in_sizes; (void)n_in; (void)out_size; (void)ws_size;

    const float* x    = (const float*)d_in[0];
    const float* dict = (const float*)d_in[1];

    float* out_common = (float*)d_out;                        // B*N*D
    float* out_resid  = out_common + (size_t)ROWS * DD;       // B*N*D
    float* aux_out    = out_common + (size_t)2 * ROWS * DD;   // 1

    float* ws         = (float*)d_ws;
    float* scores_ws  = ws + WS_SCORES;
    float* colsum_pt  = ws + WS_COLPART;
    float* f_ws       = ws + WS_F;
    float* ortho_ws   = ws + WS_ORTHO;
    float* kl_part    = ws + WS_KLPART;

    csc_main_kernel<<<GRID1, BLOCK1, 0, stream>>>(x, dict, out_common, out_resid,
                                                  scores_ws, colsum_pt);
    csc_reduce_kernel<<<1, 256, 0, stream>>>(colsum_pt, dict, f_ws, ortho_ws);
    csc_kl_kernel<<<GRIDC, BLOCKC, 0, stream>>>(scores_ws, f_ws, kl_part);
    csc_final_kernel<<<1, 256, 0, stream>>>(kl_part, ortho_ws, aux_out);
}